// FeedForwardQuantum_65481071410500
// MI455X (gfx1250) — compile-verified
//
#include <hip/hip_runtime.h>

typedef float v2f __attribute__((ext_vector_type(2)));
typedef float v8f __attribute__((ext_vector_type(8)));

#define NQ   16384   // B*T rows
#define CDIM 256
#define FFN  1024
#define DIMQ 16      // 2^4 state amplitudes
#define ROWP 1028    // padded LDS row stride in floats (1024 + 4 -> bank offset 4/row)
#define MT   4       // 16-row tiles per block in the fused MLP kernel

// ---------------------------------------------------------------------------
// Quantum circuit simulation: one thread per sample, state in registers.
// Wire w corresponds to bit (3-w) of the 4-bit amplitude index (bv = 8 >> w).
// ---------------------------------------------------------------------------
__device__ __forceinline__ void g_rx(float re[DIMQ], float im[DIMQ], int bv,
                                     float c, float s) {
#pragma unroll
  for (int i = 0; i < DIMQ; ++i) {
    if (i & bv) continue;
    int j = i | bv;
    float r0 = re[i], i0 = im[i], r1 = re[j], i1 = im[j];
    re[i] = c * r0 + s * i1;
    im[i] = c * i0 - s * r1;
    re[j] = s * i0 + c * r1;
    im[j] = c * i1 - s * r0;
  }
}

__device__ __forceinline__ void g_ry(float re[DIMQ], float im[DIMQ], int bv,
                                     float c, float s) {
#pragma unroll
  for (int i = 0; i < DIMQ; ++i) {
    if (i & bv) continue;
    int j = i | bv;
    float r0 = re[i], i0 = im[i], r1 = re[j], i1 = im[j];
    re[i] = c * r0 - s * r1;
    im[i] = c * i0 - s * i1;
    re[j] = s * r0 + c * r1;
    im[j] = s * i0 + c * i1;
  }
}

__device__ __forceinline__ void g_rz(float re[DIMQ], float im[DIMQ], int bv,
                                     float c, float s) {
#pragma unroll
  for (int i = 0; i < DIMQ; ++i) {
    if (i & bv) continue;
    int j = i | bv;
    float r0 = re[i], i0 = im[i], r1 = re[j], i1 = im[j];
    // bit==0 amplitude *= e^{-i t/2} = (c, -s); bit==1 amplitude *= (c, +s)
    re[i] = c * r0 + s * i0;
    im[i] = c * i0 - s * r0;
    re[j] = c * r1 - s * i1;
    im[j] = c * i1 + s * r1;
  }
}

__device__ __forceinline__ void g_cnot(float re[DIMQ], float im[DIMQ], int cb,
                                       int tb) {
#pragma unroll
  for (int i = 0; i < DIMQ; ++i) {
    if ((i & cb) && !(i & tb)) {
      int j = i | tb;
      float tr = re[i], ti = im[i];
      re[i] = re[j]; im[i] = im[j];
      re[j] = tr;    im[j] = ti;
    }
  }
}

__global__ __launch_bounds__(256) void qsim_kernel(
    const float* __restrict__ x, const float* __restrict__ rand_params,
    const float* __restrict__ rx_theta, const float* __restrict__ ry_theta,
    float* __restrict__ q) {
  int row = blockIdx.x * blockDim.x + threadIdx.x;
  if (row >= NQ) return;

  const float4 ang = *(const float4*)(x + (size_t)row * CDIM);
  const float angles[4] = {ang.x, ang.y, ang.z, ang.w};

  float re[DIMQ], im[DIMQ];
#pragma unroll
  for (int i = 0; i < DIMQ; ++i) { re[i] = 0.f; im[i] = 0.f; }
  re[0] = 1.f;

  // Encoder: RY(x_w) on wire w
#pragma unroll
  for (int w = 0; w < 4; ++w) {
    float s, c;
    sincosf(angles[w] * 0.5f, &s, &c);
    g_ry(re, im, 8 >> w, c, s);
  }

  // Random layer: gate type i%3 (RX,RY,RZ), wire (7*i)%4, CNOTs interleaved
  constexpr int GT[20] = {0,1,2,0,1,2,0,1,2,0,1,2,0,1,2,0,1,2,0,1};
  constexpr int GW[20] = {0,3,2,1,0,3,2,1,0,3,2,1,0,3,2,1,0,3,2,1};
#pragma unroll
  for (int i = 0; i < 20; ++i) {
    float s, c;
    sincosf(rand_params[i] * 0.5f, &s, &c);
    int bv = 8 >> GW[i];
    if (GT[i] == 0)      g_rx(re, im, bv, c, s);
    else if (GT[i] == 1) g_ry(re, im, bv, c, s);
    else                 g_rz(re, im, bv, c, s);
    if (i == 4)  g_cnot(re, im, 8 >> 0, 8 >> 1);
    if (i == 9)  g_cnot(re, im, 8 >> 1, 8 >> 2);
    if (i == 14) g_cnot(re, im, 8 >> 2, 8 >> 3);
    if (i == 19) g_cnot(re, im, 8 >> 3, 8 >> 0);
  }

  // Shared RX(rx_theta) then RY(ry_theta) on each wire
  float sx, cx, sy, cy;
  sincosf(rx_theta[0] * 0.5f, &sx, &cx);
  sincosf(ry_theta[0] * 0.5f, &sy, &cy);
#pragma unroll
  for (int w = 0; w < 4; ++w) {
    g_rx(re, im, 8 >> w, cx, sx);
    g_ry(re, im, 8 >> w, cy, sy);
  }

  // MeasureAll PauliZ
  float4 out;
  float e[4];
#pragma unroll
  for (int w = 0; w < 4; ++w) {
    int bv = 8 >> w;
    float acc = 0.f;
#pragma unroll
    for (int i = 0; i < DIMQ; ++i) {
      float p = re[i] * re[i] + im[i] * im[i];
      acc += (i & bv) ? -p : p;
    }
    e[w] = acc;
  }
  out.x = e[0]; out.y = e[1]; out.z = e[2]; out.w = e[3];
  *(float4*)(q + (size_t)row * 4) = out;
}

// ---------------------------------------------------------------------------
// Fused MLP: h = relu(q @ W1^T + b1) staged in LDS, out = h @ W2^T + b2.
// All matrix math via V_WMMA_F32_16X16X4_F32 (exact f32, K=4 native shape).
// Block = 256 threads (8 waves), owns MT*16 = 64 rows.
// ---------------------------------------------------------------------------
__global__ __launch_bounds__(256) void mlp_kernel(
    const float* __restrict__ q, const float* __restrict__ W1,
    const float* __restrict__ b1, const float* __restrict__ W2,
    const float* __restrict__ b2, float* __restrict__ out) {
  extern __shared__ float hsm[];  // [MT*16][ROWP]

  const int lane = threadIdx.x & 31;
  const int wid  = threadIdx.x >> 5;     // 0..7
  const int l16  = lane & 15;
  const int hi   = lane >> 4;            // 0: K=0..1 / M=r, 1: K=2..3 / M=r+8
  const int block_row = blockIdx.x * (MT * 16);

  // ---- Phase 1: h[64][1024] = relu(q @ W1^T + b1) into LDS --------------
  {
    const int mt = wid >> 1;                       // 2 waves per row-tile
    const int m0 = block_row + mt * 16;
    // A: 16x4 tile of q, reused for this wave's 32 of the row-tile's 64 column tiles
    v2f a = *(const v2f*)(q + (size_t)(m0 + l16) * 4 + hi * 2);
    const int ntbase = (wid & 1) * 32;
    for (int t = 0; t < 32; ++t) {
      const int n0 = (ntbase + t) * 16;
      v2f b = *(const v2f*)(W1 + (size_t)(n0 + l16) * 4 + hi * 2);
      v8f c = {0.f, 0.f, 0.f, 0.f, 0.f, 0.f, 0.f, 0.f};
      c = __builtin_amdgcn_wmma_f32_16x16x4_f32(false, a, false, b,
                                                (short)0, c, false, false);
      const float bias = b1[n0 + l16];
      float* hp = hsm + (size_t)(mt * 16 + hi * 8) * ROWP + n0 + l16;
#pragma unroll
      for (int r = 0; r < 8; ++r) {
        float v = c[r] + bias;
        hp[(size_t)r * ROWP] = v > 0.f ? v : 0.f;
      }
    }
  }
  __syncthreads();

  // ---- Phase 2: out[64][256] = h @ W2^T + b2 ----------------------------
#pragma unroll
  for (int nti = 0; nti < 2; ++nti) {
    const int c0 = (wid * 2 + nti) * 16;
    v8f acc[MT];
#pragma unroll
    for (int mt = 0; mt < MT; ++mt)
      acc[mt] = (v8f){0.f, 0.f, 0.f, 0.f, 0.f, 0.f, 0.f, 0.f};

    for (int k = 0; k < FFN / 4; ++k) {
      const int f0 = k * 4;
      // B: 4x16 tile of W2^T, loaded once, reused across MT row-tiles
      v2f b = *(const v2f*)(W2 + (size_t)(c0 + l16) * FFN + f0 + hi * 2);
#pragma unroll
      for (int mt = 0; mt < MT; ++mt) {
        v2f a = *(const v2f*)(hsm + (size_t)(mt * 16 + l16) * ROWP + f0 + hi * 2);
        acc[mt] = __builtin_amdgcn_wmma_f32_16x16x4_f32(
            false, a, false, b, (short)0, acc[mt], false, false);
      }
    }

    const float bias = b2[c0 + l16];
#pragma unroll
    for (int mt = 0; mt < MT; ++mt) {
      const int mrow = block_row + mt * 16 + hi * 8;
      float* op = out + (size_t)mrow * CDIM + c0 + l16;
#pragma unroll
      for (int r = 0; r < 8; ++r)
        op[(size_t)r * CDIM] = acc[mt][r] + bias;
    }
  }
}

// ---------------------------------------------------------------------------
extern "C" void kernel_launch(void* const* d_in, const int* in_sizes, int n_in,
                              void* d_out, int out_size, void* d_ws,
                              size_t ws_size, hipStream_t stream) {
  const float* x    = (const float*)d_in[0];
  const float* W1   = (const float*)d_in[1];
  const float* b1   = (const float*)d_in[2];
  const float* W2   = (const float*)d_in[3];
  const float* b2   = (const float*)d_in[4];
  const float* rp   = (const float*)d_in[5];
  const float* rxth = (const float*)d_in[6];
  const float* ryth = (const float*)d_in[7];
  float* out = (float*)d_out;
  float* q = (float*)d_ws;  // [NQ][4] f32 = 256 KB scratch

  qsim_kernel<<<NQ / 256, 256, 0, stream>>>(x, rp, rxth, ryth, q);

  const size_t lds_bytes = (size_t)(MT * 16) * ROWP * sizeof(float);  // ~263 KB
  mlp_kernel<<<NQ / (MT * 16), 256, lds_bytes, stream>>>(q, W1, b1, W2, b2, out);
}